// SSIM3D_14894946582822
// MI455X (gfx1250) — compile-verified
//
#include <hip/hip_runtime.h>
#include <hip/hip_bf16.h>

// ---------------------------------------------------------------------------
// 3D SSIM on MI455X (gfx1250), fp32, separable 11-tap Gaussian.
// Roofline: ~3 GFLOP vs ~0.5 GB traffic -> bandwidth bound at 23.3 TB/s.
// W/H convolutions are banded-matrix GEMMs on V_WMMA_F32_16X16X4_F32 (fp32
// matrix pipe); D convolution + SSIM map is a streaming register-ring kernel
// with global_prefetch; final reduction is deterministic (fixed order).
// Workspace requirement: 5 * 4*128^3 * 4B + 1KB ~= 167.8 MB.
// ---------------------------------------------------------------------------

typedef float v2f __attribute__((ext_vector_type(2)));
typedef float v8f __attribute__((ext_vector_type(8)));

#define SSIM_C1 (0.01f * 0.01f)
#define SSIM_C2 (0.03f * 0.03f)

// Normalized 1D Gaussian, size 11, sigma 1.5 (matches reference numerics).
__constant__ float c_gw[11] = {
    0.00102838f, 0.00759876f, 0.03600078f, 0.10936058f, 0.21300580f,
    0.26601210f, 0.21300580f, 0.10936058f, 0.03600078f, 0.00759876f,
    0.00102838f};

__device__ __forceinline__ float fieldval(int f, float a, float b) {
  switch (f) {
    case 0:  return a;        // img1        -> mu1
    case 1:  return b;        // img2        -> mu2
    case 2:  return a * a;    // img1^2      -> E[x^2]
    case 3:  return b * b;    // img2^2      -> E[y^2]
    default: return a * b;    // img1*img2   -> E[xy]
  }
}

// ---------------------------------------------------------------------------
// Kernel 1: fused H+W separable convolution of the 5 SSIM fields, per (n,d)
// slice, via WMMA f32 16x16x4 banded GEMMs. One wave = one 16x16 output tile.
// Grid: 4*128*64 tiles / 8 waves per block = 4096 blocks of 256 threads.
// ---------------------------------------------------------------------------
__global__ __launch_bounds__(256) void ssim3d_hw_wmma(
    const float* __restrict__ img1, const float* __restrict__ img2,
    float* __restrict__ fout) {
  // Per-wave LDS partitions (8 waves). Total static LDS = 63,616 B (< 64KB).
  __shared__ float sA[8][26][28];   // img1 halo (rows 0..25, cols 0..25; 26/27 zero)
  __shared__ float sB[8][26][28];   // img2 halo
  __shared__ float sT[8][28][17];   // W-conv result (rows 0..25 valid, 26/27 zero)
  __shared__ float sBand[28][16];   // Band[k][j] = g[k-j], shared by block

  const int lane = threadIdx.x & 31;
  const int wid  = threadIdx.x >> 5;

  // Build the banded weight matrix once per block.
  for (int idx = threadIdx.x; idx < 28 * 16; idx += 256) {
    const int k = idx >> 4, j = idx & 15;
    const int b = k - j;
    sBand[k][j] = (b >= 0 && b < 11) ? c_gw[b] : 0.0f;
  }

  const int tile  = blockIdx.x * 8 + wid;    // 0 .. 32767
  const int tw    = tile & 7;
  const int th    = (tile >> 3) & 7;
  const int slice = tile >> 6;               // n*128 + d
  const int h0 = th * 16, w0 = tw * 16;
  const size_t sliceBase = (size_t)slice * (128 * 128);

  // Cooperative coalesced load of the (26 x 26) halo, zero-padded (SAME conv
  // uses zero padding). Cols 26,27 are K-padding zeros.
  for (int idx = lane; idx < 26 * 28; idx += 32) {
    const int r = idx / 28, c = idx % 28;
    const int h = h0 - 5 + r, w = w0 - 5 + c;
    float v1 = 0.0f, v2 = 0.0f;
    if (c < 26 && (unsigned)h < 128u && (unsigned)w < 128u) {
      const size_t g = sliceBase + (size_t)h * 128 + w;
      v1 = img1[g];
      v2 = img2[g];
    }
    sA[wid][r][c] = v1;
    sB[wid][r][c] = v2;
  }
  // Zero K-padding rows 26,27 of the transpose buffer (stay zero all fields).
  for (int idx = lane; idx < 2 * 17; idx += 32)
    sT[wid][26 + idx / 17][idx % 17] = 0.0f;
  __syncthreads();

  // WMMA f32 16x16x4 operand lane mapping (ISA 7.12.2):
  //  A: lanes 0-15 rows M=0..15; VGPR0 = K, VGPR1 = K+1; hi lane-half K+=2.
  //  B: VGPR0 = row K (lanes 0-15) / row K+2 (lanes 16-31); VGPR1 = K+1 / K+3.
  //  C: VGPR v = row v (lanes 0-15) / row v+8 (lanes 16-31).
  const int j    = lane & 15;           // A row m  /  B,C column n
  const int koff = (lane & 16) ? 2 : 0; // K offset for high lane-half
  const int rAdd = (lane & 16) ? 8 : 0; // C row offset for high lane-half

  const size_t vol = 4ull * 128 * 128 * 128;

  for (int f = 0; f < 5; ++f) {
    // ---- W-direction conv:  T1[26x16] = Field[26x26(+pad)] x Band[28x16]
    for (int rg = 0; rg < 2; ++rg) {
      const int rb = rg * 16;
      const int rr = rb + j;                    // data row (may be >= 26)
      // Branch-free OOB handling: clamp the row address (loop-invariant
      // cndmask, keeps EXEC intact and lets ds_loads batch/pipeline) and
      // kill the contribution with a loop-invariant 0/1 mask multiply.
      const int   rcl  = (rr < 26) ? rr : 0;
      const float mrow = (rr < 26) ? 1.0f : 0.0f;
      v8f acc = {0.f, 0.f, 0.f, 0.f, 0.f, 0.f, 0.f, 0.f};
      #pragma unroll
      for (int kc = 0; kc < 7; ++kc) {
        const int k0 = kc * 4 + koff;
        const float x1a = sA[wid][rcl][k0];
        const float x2a = sB[wid][rcl][k0];
        const float x1b = sA[wid][rcl][k0 + 1];
        const float x2b = sB[wid][rcl][k0 + 1];
        v2f A, B;
        A.x = fieldval(f, x1a, x2a) * mrow;
        A.y = fieldval(f, x1b, x2b) * mrow;
        B.x = sBand[k0][j];
        B.y = sBand[k0 + 1][j];
        acc = __builtin_amdgcn_wmma_f32_16x16x4_f32(
            false, A, false, B, (short)0, acc, false, false);
      }
      #pragma unroll
      for (int v = 0; v < 8; ++v) {
        const int row = rb + v + rAdd;
        if (row < 26) sT[wid][row][j] = acc[v];   // masked stores, post-chain
      }
    }
    __syncthreads();

    // ---- H-direction conv:  Out[16x16] = Band'[16x28] x T1[28x16]
    //      Band'[m][k] = g[k-m] = sBand[k][m]
    v8f acc2 = {0.f, 0.f, 0.f, 0.f, 0.f, 0.f, 0.f, 0.f};
    #pragma unroll
    for (int kc = 0; kc < 7; ++kc) {
      const int k0 = kc * 4 + koff;
      v2f A, B;
      A.x = sBand[k0][j];
      A.y = sBand[k0 + 1][j];
      B.x = sT[wid][k0][j];
      B.y = sT[wid][k0 + 1][j];
      acc2 = __builtin_amdgcn_wmma_f32_16x16x4_f32(
          false, A, false, B, (short)0, acc2, false, false);
    }

    float* dst = fout + (size_t)f * vol + sliceBase;
    #pragma unroll
    for (int v = 0; v < 8; ++v) {
      const int hh = h0 + v + rAdd;
      dst[(size_t)hh * 128 + (w0 + j)] = acc2[v];
    }
    __syncthreads();  // sT is reused by the next field
  }
}

// ---------------------------------------------------------------------------
// Kernel 2: D-direction 11-tap conv (register shift-ring, static indexing
// only -> no scratch), fused SSIM map + per-block sum. One thread per
// (n,h,w) column streaming over d. 65536 columns -> 256 blocks x 256.
// ---------------------------------------------------------------------------
__global__ __launch_bounds__(256) void ssim3d_dconv_ssim(
    const float* __restrict__ F, float* __restrict__ blockSums) {
  const size_t vol = 4ull * 128 * 128 * 128;
  const int col = blockIdx.x * 256 + threadIdx.x;  // 0 .. 65535
  const int w = col & 127;
  const int h = (col >> 7) & 127;
  const int n = col >> 14;
  const size_t base = (size_t)n * (128 * 128 * 128) + (size_t)h * 128 + w;

  // ring[f][t] holds plane (dout - 5 + t); planes outside [0,128) are zero.
  float ring[5][11];
  #pragma unroll
  for (int f = 0; f < 5; ++f)
    #pragma unroll
    for (int t = 0; t < 11; ++t) ring[f][t] = 0.0f;

  #pragma unroll
  for (int p = 0; p < 6; ++p) {  // preload planes 0..5 into t = 5..10
    const size_t idx = base + (size_t)p * (128 * 128);
    #pragma unroll
    for (int f = 0; f < 5; ++f) ring[f][5 + p] = F[(size_t)f * vol + idx];
  }

  float lsum = 0.0f;
  for (int dout = 0; dout < 128; ++dout) {
    float c0 = 0.f, c1 = 0.f, c2 = 0.f, c3 = 0.f, c4 = 0.f;
    #pragma unroll
    for (int t = 0; t < 11; ++t) {
      const float wgt = c_gw[t];
      c0 = fmaf(wgt, ring[0][t], c0);
      c1 = fmaf(wgt, ring[1][t], c1);
      c2 = fmaf(wgt, ring[2][t], c2);
      c3 = fmaf(wgt, ring[3][t], c3);
      c4 = fmaf(wgt, ring[4][t], c4);
    }
    const float mu1 = c0, mu2 = c1;
    const float mu1sq = mu1 * mu1, mu2sq = mu2 * mu2, mu12 = mu1 * mu2;
    const float s1 = c2 - mu1sq, s2 = c3 - mu2sq, s12 = c4 - mu12;
    const float num = (2.0f * mu12 + SSIM_C1) * (2.0f * s12 + SSIM_C2);
    const float den = (mu1sq + mu2sq + SSIM_C1) * (s1 + s2 + SSIM_C2);
    lsum += num / den;

    // shift the ring and stream in plane dout+6
    #pragma unroll
    for (int f = 0; f < 5; ++f)
      #pragma unroll
      for (int t = 0; t < 10; ++t) ring[f][t] = ring[f][t + 1];
    const int dn = dout + 6;
    if (dn < 128) {
      const size_t idx = base + (size_t)dn * (128 * 128);
      #pragma unroll
      for (int f = 0; f < 5; ++f) ring[f][10] = F[(size_t)f * vol + idx];
    } else {
      #pragma unroll
      for (int f = 0; f < 5; ++f) ring[f][10] = 0.0f;
    }
    // Prefetch ~8 planes ahead to hide HBM/L2 latency on this streaming
    // kernel (lowers to global_prefetch_b8 on gfx1250).
    const int dp = dout + 14;
    if (dp < 128) {
      const size_t pidx = base + (size_t)dp * (128 * 128);
      #pragma unroll
      for (int f = 0; f < 5; ++f)
        __builtin_prefetch(&F[(size_t)f * vol + pidx], 0, 3);
    }
  }

  // Block reduction (fixed order within block).
  __shared__ float red[256];
  red[threadIdx.x] = lsum;
  __syncthreads();
  #pragma unroll
  for (int s = 128; s > 0; s >>= 1) {
    if (threadIdx.x < s) red[threadIdx.x] += red[threadIdx.x + s];
    __syncthreads();
  }
  if (threadIdx.x == 0) blockSums[blockIdx.x] = red[0];
}

// ---------------------------------------------------------------------------
// Kernel 3: deterministic final reduction of 256 block sums -> mean.
// ---------------------------------------------------------------------------
__global__ __launch_bounds__(256) void ssim3d_reduce(
    const float* __restrict__ blockSums, float* __restrict__ out) {
  __shared__ float red[256];
  red[threadIdx.x] = blockSums[threadIdx.x];
  __syncthreads();
  #pragma unroll
  for (int s = 128; s > 0; s >>= 1) {
    if (threadIdx.x < s) red[threadIdx.x] += red[threadIdx.x + s];
    __syncthreads();
  }
  if (threadIdx.x == 0) out[0] = red[0] * (1.0f / 8388608.0f);
}

extern "C" void kernel_launch(void* const* d_in, const int* in_sizes, int n_in,
                              void* d_out, int out_size, void* d_ws,
                              size_t ws_size, hipStream_t stream) {
  (void)in_sizes; (void)n_in; (void)out_size; (void)ws_size;
  const float* img1 = (const float*)d_in[0];
  const float* img2 = (const float*)d_in[1];
  // d_in[2] is the precomputed 11^3 window; we use the identical separable
  // weights from constant memory instead.

  const size_t vol = 4ull * 128 * 128 * 128;      // 8,388,608 elems per field
  float* fields = (float*)d_ws;                   // 5 * vol floats (167.8 MB)
  float* bsums  = fields + 5 * vol;               // 256 floats

  ssim3d_hw_wmma<<<4096, 256, 0, stream>>>(img1, img2, fields);
  ssim3d_dconv_ssim<<<256, 256, 0, stream>>>(fields, bsums);
  ssim3d_reduce<<<1, 256, 0, stream>>>(bsums, (float*)d_out);
}